// DecoderGAT_40089224741038
// MI455X (gfx1250) — compile-verified
//
#include <hip/hip_runtime.h>
#include <hip/hip_bf16.h>
#include <math.h>

typedef float v2f __attribute__((ext_vector_type(2)));
typedef float v8f __attribute__((ext_vector_type(8)));

#define IN_DIM 128
#define HEADS 4
#define HEAD_DIM 32
#define NEG_SLOPE 0.2f

// ---------------------------------------------------------------------------
// K1: xh = x @ W via V_WMMA_F32_16X16X4_F32. One block = 8 waves; wave w
// computes the 16x16 tile at (rowTile=blockIdx.x, colTile=w). K loop: 32
// steps of 4, unrolled 8x.
// A layout (16x4 f32): lanes 0-15 -> M=lane, VGPR0=K0,VGPR1=K1;
//                      lanes 16-31 -> M=lane-16, VGPR0=K2,VGPR1=K3.
// B layout (4x16 f32): lane%16 -> N; VGPR0=K0(/K2 hi-half), VGPR1=K1(/K3).
// C/D: VGPR i holds M=i (lanes 0-15) / M=i+8 (lanes 16-31), N=lane%16.
// Row index is CLAMPED (not predicated) so EXEC stays all-1s: no exec
// save/restore in the hot loop, loads can clause.
// ---------------------------------------------------------------------------
__global__ __launch_bounds__(256) void gat_gemm_wmma(
    const float* __restrict__ x, const float* __restrict__ W,
    float* __restrict__ xh, int N)
{
  const int lane  = threadIdx.x & 31;
  const int wave  = threadIdx.x >> 5;        // 0..7 column tile
  const int row0  = blockIdx.x * 16;
  const int col0  = wave * 16;
  const int mrow  = lane & 15;
  const int khalf = (lane >> 4) * 2;         // 0 or 2

  int r = row0 + mrow;
  if (r > N - 1) r = N - 1;                  // clamp: always-valid loads
  const float* xrow = x + (size_t)r * IN_DIM;
  const int  bcol = col0 + mrow;

  v8f acc = {};
  #pragma unroll 8
  for (int k = 0; k < IN_DIM; k += 4) {
    v2f a = *(const v2f*)(xrow + k + khalf);                 // b64 load
    v2f b;
    b.x = W[(size_t)(k + khalf)     * IN_DIM + bcol];
    b.y = W[(size_t)(k + khalf + 1) * IN_DIM + bcol];
    acc = __builtin_amdgcn_wmma_f32_16x16x4_f32(
        /*neg_a=*/false, a, /*neg_b=*/false, b,
        /*c_mod=*/(short)0, acc, /*reuse_a=*/false, /*reuse_b=*/false);
  }

  const int srow0 = row0 + ((lane >> 4) << 3);   // +0 or +8
  const int scol  = col0 + mrow;
  #pragma unroll
  for (int i = 0; i < 8; ++i) {
    const int rr = srow0 + i;
    if (rr < N) xh[(size_t)rr * IN_DIM + scol] = acc[i];
  }
}

// ---------------------------------------------------------------------------
// K2: per-node attention coefficients. 128 threads/node; head h = one wave32,
// reduce C=32 products with shfl.
// ---------------------------------------------------------------------------
__global__ __launch_bounds__(128) void gat_attn_coeff(
    const float* __restrict__ xh,
    const float* __restrict__ att_src, const float* __restrict__ att_dst,
    float* __restrict__ a_src, float* __restrict__ a_dst, int N)
{
  const int node = blockIdx.x;
  if (node >= N) return;
  const int l = threadIdx.x;          // 0..127 == h*32 + c
  const int h = l >> 5;
  const int c = l & 31;
  const float v = xh[(size_t)node * IN_DIM + l];
  float s = v * att_src[l];
  float d = v * att_dst[l];
  #pragma unroll
  for (int off = 16; off > 0; off >>= 1) {
    s += __shfl_down(s, off, 32);
    d += __shfl_down(d, off, 32);
  }
  if (c == 0) {
    a_src[node * HEADS + h] = s;
    a_dst[node * HEADS + h] = d;
  }
}

// ---------------------------------------------------------------------------
// Init: m = -inf, denom = 0, loss accumulators = 0
// ---------------------------------------------------------------------------
__global__ void gat_init(float* __restrict__ m, float* __restrict__ denom,
                         float* __restrict__ loss, int NH)
{
  const int i = blockIdx.x * blockDim.x + threadIdx.x;
  if (i < NH) { m[i] = -INFINITY; denom[i] = 0.f; }
  if (i == 0) { loss[0] = 0.f; loss[1] = 0.f; }
}

// ---------------------------------------------------------------------------
// K3a: per-(edge,head) raw score + leaky relu + segment max via float atomic
// ---------------------------------------------------------------------------
__global__ __launch_bounds__(256) void gat_edge_score(
    const int* __restrict__ es, const int* __restrict__ ed,
    const float* __restrict__ a_src, const float* __restrict__ a_dst,
    float* __restrict__ ew, float* __restrict__ m, int EH)
{
  const int i = blockIdx.x * blockDim.x + threadIdx.x;
  if (i >= EH) return;
  const int e = i >> 2, h = i & 3;
  const int s = es[e], d = ed[e];
  float v = a_src[s * HEADS + h] + a_dst[d * HEADS + h];
  v = (v > 0.f) ? v : NEG_SLOPE * v;
  ew[i] = v;
  atomicMax(&m[d * HEADS + h], v);
}

// ---------------------------------------------------------------------------
// K3b: m = where(isfinite(m), m, 0)
// ---------------------------------------------------------------------------
__global__ void gat_fix_m(float* __restrict__ m, int NH)
{
  const int i = blockIdx.x * blockDim.x + threadIdx.x;
  if (i < NH) {
    const float v = m[i];
    if (!(v > -INFINITY && v < INFINITY)) m[i] = 0.f;
  }
}

// ---------------------------------------------------------------------------
// K3c: ex = exp(e - m[dst]); denom[dst] += ex
// ---------------------------------------------------------------------------
__global__ __launch_bounds__(256) void gat_edge_exp(
    const int* __restrict__ ed, const float* __restrict__ m,
    float* __restrict__ ew, float* __restrict__ denom, int EH)
{
  const int i = blockIdx.x * blockDim.x + threadIdx.x;
  if (i >= EH) return;
  const int e = i >> 2, h = i & 3;
  const int d = ed[e];
  const float ex = expf(ew[i] - m[d * HEADS + h]);
  ew[i] = ex;
  atomicAdd(&denom[d * HEADS + h], ex);
}

// ---------------------------------------------------------------------------
// K3d: weighted scatter: out[dst, l] += xh[src, l] * alpha(e, h)
// One thread per (edge, feature-lane). 128 lanes per edge.
// ---------------------------------------------------------------------------
__global__ __launch_bounds__(256) void gat_scatter(
    const int* __restrict__ es, const int* __restrict__ ed,
    const float* __restrict__ ew, const float* __restrict__ denom,
    const float* __restrict__ xh, float* __restrict__ out, long long total)
{
  const long long i = (long long)blockIdx.x * blockDim.x + threadIdx.x;
  if (i >= total) return;
  const int e = (int)(i >> 7);
  const int l = (int)(i & 127);
  const int h = l >> 5;
  const int s = es[e], d = ed[e];
  const float alpha = ew[e * HEADS + h] / (denom[d * HEADS + h] + 1e-16f);
  atomicAdd(&out[(size_t)d * IN_DIM + l], xh[(size_t)s * IN_DIM + l] * alpha);
}

// ---------------------------------------------------------------------------
// K4: logits = out + bias (in place), masked CE accumulation.
// One 128-thread block per node; LDS max/sum reductions across 4 waves.
// ---------------------------------------------------------------------------
__global__ __launch_bounds__(128) void gat_bias_loss(
    float* __restrict__ logits, const float* __restrict__ bias,
    const int* __restrict__ targets, float* __restrict__ loss, int N)
{
  const int n = blockIdx.x;
  if (n >= N) return;
  const int l = threadIdx.x;
  const float v = logits[(size_t)n * IN_DIM + l] + bias[l];
  logits[(size_t)n * IN_DIM + l] = v;

  __shared__ float sm[128];
  sm[l] = v; __syncthreads();
  #pragma unroll
  for (int off = 64; off > 0; off >>= 1) {
    if (l < off) sm[l] = fmaxf(sm[l], sm[l + off]);
    __syncthreads();
  }
  const float mx = sm[0]; __syncthreads();
  sm[l] = expf(v - mx); __syncthreads();
  #pragma unroll
  for (int off = 64; off > 0; off >>= 1) {
    if (l < off) sm[l] += sm[l + off];
    __syncthreads();
  }
  const float lse = mx + logf(sm[0]);

  const int t = targets[n];
  if (t >= 0 && l == t) atomicAdd(&loss[0], lse - v);   // nll = lse - logit_t
  if (t >= 0 && l == 0) atomicAdd(&loss[1], 1.0f);
}

__global__ void gat_finalize(const float* __restrict__ loss,
                             float* __restrict__ out_loss)
{
  if (blockIdx.x == 0 && threadIdx.x == 0)
    out_loss[0] = loss[0] / fmaxf(loss[1], 1.0f);
}

// ---------------------------------------------------------------------------
extern "C" void kernel_launch(void* const* d_in, const int* in_sizes, int n_in,
                              void* d_out, int out_size, void* d_ws, size_t ws_size,
                              hipStream_t stream)
{
  const float* x        = (const float*)d_in[0];
  const float* W        = (const float*)d_in[1];
  const float* att_src  = (const float*)d_in[2];
  const float* att_dst  = (const float*)d_in[3];
  const float* bias     = (const float*)d_in[4];
  const int*   edge_src = (const int*)d_in[5];
  const int*   edge_dst = (const int*)d_in[6];
  const int*   targets  = (const int*)d_in[7];
  float* out = (float*)d_out;

  const int N  = in_sizes[0] / IN_DIM;
  const int E  = in_sizes[5];
  const int NH = N * HEADS;
  const int EH = E * HEADS;

  // workspace layout (floats)
  float* ws     = (float*)d_ws;
  float* xh     = ws;                               // N*128
  float* a_src  = xh    + (size_t)N * IN_DIM;       // N*4
  float* a_dst  = a_src + NH;                       // N*4
  float* m      = a_dst + NH;                       // N*4
  float* denom  = m     + NH;                       // N*4
  float* ew     = denom + NH;                       // E*4
  float* loss   = ew    + EH;                       // 2

  // zero the logits accumulator region of d_out
  hipMemsetAsync(out, 0, (size_t)N * IN_DIM * sizeof(float), stream);
  gat_init<<<(NH + 255) / 256, 256, 0, stream>>>(m, denom, loss, NH);

  gat_gemm_wmma<<<(N + 15) / 16, 256, 0, stream>>>(x, W, xh, N);
  gat_attn_coeff<<<N, 128, 0, stream>>>(xh, att_src, att_dst, a_src, a_dst, N);

  gat_edge_score<<<(EH + 255) / 256, 256, 0, stream>>>(edge_src, edge_dst,
                                                       a_src, a_dst, ew, m, EH);
  gat_fix_m<<<(NH + 255) / 256, 256, 0, stream>>>(m, NH);
  gat_edge_exp<<<(EH + 255) / 256, 256, 0, stream>>>(edge_dst, m, ew, denom, EH);

  const long long total = (long long)E * IN_DIM;
  gat_scatter<<<(int)((total + 255) / 256), 256, 0, stream>>>(
      edge_src, edge_dst, ew, denom, xh, out, total);

  gat_bias_loss<<<N, 128, 0, stream>>>(out, bias, targets, loss, N);
  gat_finalize<<<1, 32, 0, stream>>>(loss, out + (size_t)N * IN_DIM);
}